// SepSAttn_54434415510114
// MI455X (gfx1250) — compile-verified
//
#include <hip/hip_runtime.h>
#include <math.h>

// Problem constants: x (16,256,64,64) f32, W (513,256) f32.
constexpr int BATCH = 16;
constexpr int C     = 256;    // GEMM K
constexpr int N     = 4096;   // H*W
constexpr int M     = 513;    // 1 + 2*256 qkv rows
constexpr int D     = 256;

constexpr int NCOL   = 64;      // N columns per block
constexpr int KCH    = 16;      // K rows staged per chunk
constexpr int NCHUNK = C / KCH; // 16 chunks
constexpr int XLD    = 80;      // x LDS row stride (pad: half-waves on disjoint banks)
constexpr int WLD    = 20;      // W LDS row stride (pad: 20*l distinct mod 64)

typedef float v2f __attribute__((ext_vector_type(2)));
typedef float v8f __attribute__((ext_vector_type(8)));
typedef int   v4i __attribute__((vector_size(16)));   // matches builtin param type

#if __has_builtin(__builtin_amdgcn_global_load_async_to_lds_b128)
#define USE_ASYNC_LDS 1
#else
#define USE_ASYNC_LDS 0
#endif

__device__ __forceinline__ void wait_asynccnt0() {
#if __has_builtin(__builtin_amdgcn_s_wait_asynccnt)
    __builtin_amdgcn_s_wait_asynccnt(0);
#else
    asm volatile("s_wait_asynccnt 0x0" ::: "memory");
#endif
}

#if USE_ASYNC_LDS
__device__ __forceinline__ void async_cp16(const float* g, float* l) {
    __builtin_amdgcn_global_load_async_to_lds_b128(
        (__attribute__((address_space(1))) v4i*)g,
        (__attribute__((address_space(3))) v4i*)l, 0, 0);
}
#endif

// ---------------------------------------------------------------------------
// Phase 1a: qkv rows 0..511 = W(512x256) @ x(b,256,4096), fp32 WMMA.
// Block = 128 thr = 4 waves; block tile = 64 M-rows x 64 N-cols; wave w owns
// M-tile w, four 16x16 accumulators across the 64 columns. Operands staged
// through double-buffered LDS chunks (16 K-rows each) via async b128 copies.
// Grid: (N/64, 512/64, B)
// ---------------------------------------------------------------------------
__global__ __launch_bounds__(128)
void qkv_gemm_f32_wmma(const float* __restrict__ x,
                       const float* __restrict__ Wm,
                       float* __restrict__ qkv)
{
    __shared__ float xlds[2][KCH * XLD];   // 2 x 5120 B
    __shared__ float wlds[2][64 * WLD];    // 2 x 5120 B

    const int t    = threadIdx.x;
    const int lane = t & 31;
    const int wid  = t >> 5;
    const int lo   = lane & 15;
    const int hi   = lane >> 4;        // half-wave: K pairs (0,1) vs (2,3)

    const int mbase = blockIdx.y * 64;
    const int n0    = blockIdx.x * NCOL;
    const int b     = blockIdx.z;
    const float* __restrict__ xb = x + (size_t)b * C * N;

    // staging geometry: x chunk 16x64 -> thread (t>>3, (t&7)*8), two float4
    const int xr = t >> 3, xc = (t & 7) * 8;
    // W chunk 64x16 -> thread (t>>1, (t&1)*8), two float4
    const int wr = t >> 1, wc = (t & 1) * 8;

    v8f acc[4] = {v8f{}, v8f{}, v8f{}, v8f{}};

    // ---- prologue: stage chunk 0 ----
#if USE_ASYNC_LDS
    {
        const float* gx = xb + (size_t)xr * N + n0 + xc;
        const float* gw = Wm + (size_t)(mbase + wr) * C + wc;
        async_cp16(gx,     &xlds[0][xr * XLD + xc]);
        async_cp16(gx + 4, &xlds[0][xr * XLD + xc + 4]);
        async_cp16(gw,     &wlds[0][wr * WLD + wc]);
        async_cp16(gw + 4, &wlds[0][wr * WLD + wc + 4]);
        wait_asynccnt0();
    }
#else
    {
        const float* gx = xb + (size_t)xr * N + n0 + xc;
        const float* gw = Wm + (size_t)(mbase + wr) * C + wc;
        float4 x0 = *(const float4*)gx, x1 = *(const float4*)(gx + 4);
        float4 w0 = *(const float4*)gw, w1 = *(const float4*)(gw + 4);
        *(float4*)&xlds[0][xr * XLD + xc]     = x0;
        *(float4*)&xlds[0][xr * XLD + xc + 4] = x1;
        *(float4*)&wlds[0][wr * WLD + wc]     = w0;
        *(float4*)&wlds[0][wr * WLD + wc + 4] = w1;
    }
#endif
    __syncthreads();

    for (int ch = 0; ch < NCHUNK; ++ch) {
        const float* xl = xlds[ch & 1];
        const float* wl = wlds[ch & 1];

        // ---- stage chunk ch+1 into the other buffer ----
#if USE_ASYNC_LDS
        if (ch + 1 < NCHUNK) {
            const float* gx = xb + (size_t)((ch + 1) * KCH + xr) * N + n0 + xc;
            const float* gw = Wm + (size_t)(mbase + wr) * C + (ch + 1) * KCH + wc;
            float* lx = &xlds[(ch + 1) & 1][xr * XLD + xc];
            float* lw = &wlds[(ch + 1) & 1][wr * WLD + wc];
            async_cp16(gx, lx);  async_cp16(gx + 4, lx + 4);
            async_cp16(gw, lw);  async_cp16(gw + 4, lw + 4);
        }
#else
        float4 x0{}, x1{}, w0{}, w1{};
        if (ch + 1 < NCHUNK) {
            const float* gx = xb + (size_t)((ch + 1) * KCH + xr) * N + n0 + xc;
            const float* gw = Wm + (size_t)(mbase + wr) * C + (ch + 1) * KCH + wc;
            x0 = *(const float4*)gx; x1 = *(const float4*)(gx + 4);
            w0 = *(const float4*)gw; w1 = *(const float4*)(gw + 4);
        }
#endif

        // ---- compute: 4 k-steps of 4, 16 WMMAs per wave per chunk ----
        #pragma unroll
        for (int kk = 0; kk < KCH; kk += 4) {
            const int ka = kk + 2 * hi;   // local K row for this half-wave
            // A 16x4: lane row = wid*16+lo, K = ka, ka+1 (8B-aligned b64)
            v2f a = *(const v2f*)&wl[(wid * 16 + lo) * WLD + ka];
            #pragma unroll
            for (int j = 0; j < 4; ++j) {
                v2f bv;
                bv.x = xl[ka * XLD + 16 * j + lo];
                bv.y = xl[(ka + 1) * XLD + 16 * j + lo];
                acc[j] = __builtin_amdgcn_wmma_f32_16x16x4_f32(
                    false, a, false, bv, (short)0, acc[j], false, false);
            }
        }

#if USE_ASYNC_LDS
        if (ch + 1 < NCHUNK) wait_asynccnt0();
#else
        if (ch + 1 < NCHUNK) {
            float* lx = &xlds[(ch + 1) & 1][xr * XLD + xc];
            float* lw = &wlds[(ch + 1) & 1][wr * WLD + wc];
            *(float4*)lx = x0; *(float4*)(lx + 4) = x1;
            *(float4*)lw = w0; *(float4*)(lw + 4) = w1;
        }
#endif
        __syncthreads();
    }

    // C/D layout: VGPR r -> row (wave M-tile) + r + 8*hi, column lo.
    const int mrow = mbase + wid * 16 + 8 * hi;     // <= 511, no guard needed
    #pragma unroll
    for (int j = 0; j < 4; ++j) {
        float* dst = qkv + ((size_t)b * M + mrow) * N + n0 + 16 * j + lo;
        #pragma unroll
        for (int r = 0; r < 8; ++r)
            dst[(size_t)r * N] = acc[j][r];
    }
}

// ---------------------------------------------------------------------------
// Phase 1b: leftover row o = 512 (ragged last row of W): coalesced mat-vec.
// Grid: (N/256, B), block 256. x is L2-resident after phase 1a.
// ---------------------------------------------------------------------------
__global__ __launch_bounds__(256)
void qkv_last_row(const float* __restrict__ x,
                  const float* __restrict__ Wm,
                  float* __restrict__ qkv)
{
    const int n = blockIdx.x * 256 + threadIdx.x;
    const int b = blockIdx.y;
    const float* __restrict__ wrow = Wm + (size_t)(M - 1) * C;
    const float* __restrict__ xc   = x + (size_t)b * C * N + n;
    float s = 0.f;
    for (int k = 0; k < C; ++k) s = fmaf(wrow[k], xc[(size_t)k * N], s);
    qkv[((size_t)b * M + (M - 1)) * N + n] = s;
}

// ---------------------------------------------------------------------------
// Phase 2: scores[b,:] = softmax(qkv[b,0,:]) ; one block per batch.
// ---------------------------------------------------------------------------
__global__ __launch_bounds__(256)
void softmax_q(const float* __restrict__ qkv, float* __restrict__ scores)
{
    __shared__ float red[256];
    const int b = blockIdx.x;
    const int t = threadIdx.x;
    const float* __restrict__ q = qkv + (size_t)b * M * N;   // row o=0

    float mx = -INFINITY;
    for (int n = t; n < N; n += 256) mx = fmaxf(mx, q[n]);
    red[t] = mx; __syncthreads();
    for (int s = 128; s > 0; s >>= 1) {
        if (t < s) red[t] = fmaxf(red[t], red[t + s]);
        __syncthreads();
    }
    mx = red[0]; __syncthreads();

    float sum = 0.f;
    for (int n = t; n < N; n += 256) sum += __expf(q[n] - mx);
    red[t] = sum; __syncthreads();
    for (int s = 128; s > 0; s >>= 1) {
        if (t < s) red[t] += red[t + s];
        __syncthreads();
    }
    const float inv = 1.f / red[0];

    float* __restrict__ sc = scores + (size_t)b * N;
    for (int n = t; n < N; n += 256) sc[n] = __expf(q[n] - mx) * inv;
}

// ---------------------------------------------------------------------------
// Phase 3 (fused): ctx[b,d] = sum_n k[b,d,n]*scores[b,n];
//                  out[b,d,n] = relu(v[b,d,n]) * ctx[b,d]
// One block per (b,d) row.
// ---------------------------------------------------------------------------
__global__ __launch_bounds__(256)
void ctx_relu_out(const float* __restrict__ qkv,
                  const float* __restrict__ scores,
                  float* __restrict__ out)
{
    __shared__ float red[256];
    const int d = blockIdx.x;
    const int b = blockIdx.y;
    const int t = threadIdx.x;

    const float* __restrict__ krow = qkv + ((size_t)b * M + 1 + d) * N;
    const float* __restrict__ vrow = qkv + ((size_t)b * M + 1 + D + d) * N;
    const float* __restrict__ sc   = scores + (size_t)b * N;

    float p = 0.f;
    for (int n = t; n < N; n += 256) p += krow[n] * sc[n];
    red[t] = p; __syncthreads();
    for (int s = 128; s > 0; s >>= 1) {
        if (t < s) red[t] += red[t + s];
        __syncthreads();
    }
    const float ctx = red[0];

    float* __restrict__ orow = out + ((size_t)b * D + d) * N;
    for (int n = t; n < N; n += 256) {
        const float v = vrow[n];
        orow[n] = fmaxf(v, 0.f) * ctx;
    }
}

// ---------------------------------------------------------------------------
extern "C" void kernel_launch(void* const* d_in, const int* in_sizes, int n_in,
                              void* d_out, int out_size, void* d_ws, size_t ws_size,
                              hipStream_t stream)
{
    const float* x  = (const float*)d_in[0];   // (16, 256, 64, 64)
    const float* Wm = (const float*)d_in[1];   // (513, 256)
    float* out = (float*)d_out;                // (16, 256, 64, 64)

    // ws: qkv (16*513*4096 f32 = 134.5 MB) | scores (16*4096 f32)
    float* qkv    = (float*)d_ws;
    float* scores = qkv + (size_t)BATCH * M * N;

    (void)in_sizes; (void)n_in; (void)out_size; (void)ws_size;

    qkv_gemm_f32_wmma<<<dim3(N / NCOL, 512 / 64, BATCH), 128, 0, stream>>>(x, Wm, qkv);
    qkv_last_row<<<dim3(N / 256, BATCH), 256, 0, stream>>>(x, Wm, qkv);
    softmax_q<<<dim3(BATCH), 256, 0, stream>>>(qkv, scores);
    ctx_relu_out<<<dim3(D, BATCH), 256, 0, stream>>>(qkv, scores, out);
}